// DyHuCoG_44753559225050
// MI455X (gfx1250) — compile-verified
//
#include <hip/hip_runtime.h>

// ---------------------------------------------------------------------------
// LightGCN propagation: acc = (E0 + E1 + E2 + E3)/4 with E_{k+1} = S * E_k,
// S sparse (4.8M nnz), D = 64.
//
// Roofline (MI455X): ~0.61 GFLOP/layer vs ~2.5 GB/layer of gather+scatter
// traffic -> memory/atomic bound. Embedding (38.4 MB), both ping-pong
// buffers and the edge lists all fit in the 192 MB L2, so the working set is
// L2-resident: the right shape is coalesced 128-bit gathers + f32 L2 atomics,
// with edge metadata streamed into LDS via the CDNA5 async-to-LDS path
// (ASYNCcnt-tracked, double buffered) so it is read from global exactly once.
// WMMA is inapplicable: per-edge scalar*vector scale is not a matmul.
// ---------------------------------------------------------------------------

#define DIM_C   64
#define EPT     256      // edges per LDS tile
#define BLOCK   256      // threads per block (8 wave32)

// ---- CDNA5 async global->LDS staging (inline asm; ISA 15.18.3, opcode 96) --
__device__ __forceinline__ unsigned lds_off_u32(const void* p) {
  // generic shared pointer = {shared_aperture, lds_offset}; low 32 bits = LDS byte offset
  return (unsigned)(unsigned long long)p;
}

__device__ __forceinline__ void async_ld_b32(const void* gaddr, unsigned lds_off) {
  // VDST = LDS byte address VGPR, VADDR = 64-bit global address VGPR pair
  asm volatile("global_load_async_to_lds_b32 %0, %1, off"
               :: "v"(lds_off), "v"(gaddr) : "memory");
}

#define S_WAIT_ASYNCCNT_3() asm volatile("s_wait_asynccnt 0x3" ::: "memory")
#define S_WAIT_ASYNCCNT_0() asm volatile("s_wait_asynccnt 0x0" ::: "memory")

// ---------------------------------------------------------------------------
// SpMM: dst[r,:] += vals[e] * src[c,:]  for each edge e = (r, c)
// 16 lanes per edge, each lane owns a float4 slice of the 64-wide row.
// ---------------------------------------------------------------------------
__global__ void __launch_bounds__(BLOCK)
spmm_kernel(const float* __restrict__ src, float* __restrict__ dst,
            const float* __restrict__ vals, const int* __restrict__ rows,
            const int* __restrict__ cols, int nEdges)
{
  __shared__ int   s_rows[2][EPT];
  __shared__ int   s_cols[2][EPT];
  __shared__ float s_vals[2][EPT];

  const int tid    = (int)threadIdx.x;
  const int stride = (int)gridDim.x;
  const int nTiles = (nEdges + EPT - 1) / EPT;

  int tile = (int)blockIdx.x;
  if (tile >= nTiles) return;

  // Prime buffer 0 with the first tile (3 async b32 loads per thread).
  {
    int e  = tile * EPT + tid;
    int ec = e < nEdges ? e : 0;
    async_ld_b32(&rows[ec], lds_off_u32(&s_rows[0][tid]));
    async_ld_b32(&cols[ec], lds_off_u32(&s_cols[0][tid]));
    async_ld_b32(&vals[ec], lds_off_u32(&s_vals[0][tid]));
  }

  int buf = 0;
  for (; tile < nTiles; tile += stride) {
    const int nextTile = tile + stride;
    if (nextTile < nTiles) {
      // Prefetch next tile into the other buffer, then wait for current tile:
      // exactly the 3 just-issued async ops may remain outstanding.
      const int nb = buf ^ 1;
      int e  = nextTile * EPT + tid;
      int ec = e < nEdges ? e : 0;
      async_ld_b32(&rows[ec], lds_off_u32(&s_rows[nb][tid]));
      async_ld_b32(&cols[ec], lds_off_u32(&s_cols[nb][tid]));
      async_ld_b32(&vals[ec], lds_off_u32(&s_vals[nb][tid]));
      S_WAIT_ASYNCCNT_3();
    } else {
      S_WAIT_ASYNCCNT_0();
    }
    __syncthreads();   // all waves' async data visible in LDS

    const int base = tile * EPT;
#pragma unroll
    for (int k = 0; k < (EPT * 16) / BLOCK; ++k) {
      const int slot = k * BLOCK + tid;
      const int el   = slot >> 4;   // edge within tile (16 lanes share it)
      const int ln   = slot & 15;   // float4 slice index
      const int eg   = base + el;
      if (eg < nEdges) {
        const int   r = s_rows[buf][el];
        const int   c = s_cols[buf][el];
        const float v = s_vals[buf][el];
        const float4 x = *(const float4*)(src + (size_t)c * DIM_C + ln * 4);
        float* p = dst + (size_t)r * DIM_C + ln * 4;
        atomicAdd(p + 0, x.x * v);
        atomicAdd(p + 1, x.y * v);
        atomicAdd(p + 2, x.z * v);
        atomicAdd(p + 3, x.w * v);
      }
    }
    __syncthreads();   // LDS tile fully consumed before it is overwritten
    buf ^= 1;
  }
}

// ---------------------------------------------------------------------------
// Elementwise helpers (float4-vectorized, 16B loads/stores)
// ---------------------------------------------------------------------------
__global__ void init_kernel(const float4* __restrict__ emb, float4* __restrict__ acc,
                            float4* __restrict__ bufA, float4* __restrict__ bufB, int n4)
{
  const int i = (int)(blockIdx.x * blockDim.x + threadIdx.x);
  if (i < n4) {
    const float4 e = emb[i];
    float4 a; a.x = 0.25f * e.x; a.y = 0.25f * e.y; a.z = 0.25f * e.z; a.w = 0.25f * e.w;
    acc[i] = a;
    const float4 z = make_float4(0.f, 0.f, 0.f, 0.f);
    bufA[i] = z;
    bufB[i] = z;
  }
}

__global__ void axpy_kernel(float4* __restrict__ acc, const float4* __restrict__ x, int n4)
{
  const int i = (int)(blockIdx.x * blockDim.x + threadIdx.x);
  if (i < n4) {
    float4 a = acc[i];
    const float4 b = x[i];
    a.x += 0.25f * b.x; a.y += 0.25f * b.y; a.z += 0.25f * b.z; a.w += 0.25f * b.w;
    acc[i] = a;
  }
}

__global__ void zero_kernel(float4* __restrict__ b, int n4)
{
  const int i = (int)(blockIdx.x * blockDim.x + threadIdx.x);
  if (i < n4) b[i] = make_float4(0.f, 0.f, 0.f, 0.f);
}

// ---------------------------------------------------------------------------
extern "C" void kernel_launch(void* const* d_in, const int* in_sizes, int n_in,
                              void* d_out, int out_size, void* d_ws, size_t ws_size,
                              hipStream_t stream)
{
  const float* emb  = (const float*)d_in[0];   // [N_NODES * 64] f32
  const float* vals = (const float*)d_in[1];   // [E] f32
  const int*   rows = (const int*)d_in[2];     // [E] i32
  const int*   cols = (const int*)d_in[3];     // [E] i32

  const int nElems = in_sizes[0];              // N_NODES * 64
  const int nEdges = in_sizes[1];              // 4.8M

  float* acc  = (float*)d_out;
  float* bufA = (float*)d_ws;
  float* bufB = bufA + (size_t)nElems;

  const int n4 = nElems / 4;
  const dim3 eb(256), eg((unsigned)((n4 + 255) / 256));

  const int nTiles = (nEdges + EPT - 1) / EPT;
  const int sgrid  = nTiles < 8192 ? nTiles : 8192;   // persistent grid-stride

  // acc = 0.25*E0 ; bufA = bufB = 0 (deterministic each call)
  init_kernel<<<eg, eb, 0, stream>>>((const float4*)emb, (float4*)acc,
                                     (float4*)bufA, (float4*)bufB, n4);
  // layer 1
  spmm_kernel<<<sgrid, BLOCK, 0, stream>>>(emb, bufA, vals, rows, cols, nEdges);
  axpy_kernel<<<eg, eb, 0, stream>>>((float4*)acc, (const float4*)bufA, n4);
  // layer 2
  spmm_kernel<<<sgrid, BLOCK, 0, stream>>>(bufA, bufB, vals, rows, cols, nEdges);
  axpy_kernel<<<eg, eb, 0, stream>>>((float4*)acc, (const float4*)bufB, n4);
  // layer 3 (reuse bufA after re-zeroing)
  zero_kernel<<<eg, eb, 0, stream>>>((float4*)bufA, n4);
  spmm_kernel<<<sgrid, BLOCK, 0, stream>>>(bufB, bufA, vals, rows, cols, nEdges);
  axpy_kernel<<<eg, eb, 0, stream>>>((float4*)acc, (const float4*)bufA, n4);
}